// Basic_Net_26396869001307
// MI455X (gfx1250) — compile-verified
//
#include <hip/hip_runtime.h>
#include <math.h>

// Problem constants (from the reference)
#define F_IN 512
#define H1   64
#define H2   256

typedef __attribute__((ext_vector_type(2))) float v2f;
typedef __attribute__((ext_vector_type(8))) float v8f;

// ---------------------------------------------------------------------------
// Degree / normalization
// ---------------------------------------------------------------------------
__global__ void deg_init_k(float* deg, int n) {
    int i = blockIdx.x * blockDim.x + threadIdx.x;
    if (i < n) deg[i] = 1.0f;   // self loop
}

__global__ void deg_count_k(const int* __restrict__ dst, int e, float* deg) {
    int i = blockIdx.x * blockDim.x + threadIdx.x;
    if (i < e) atomicAdd(&deg[dst[i]], 1.0f);
}

__global__ void deg_rsqrt_k(float* deg, int n) {
    int i = blockIdx.x * blockDim.x + threadIdx.x;
    if (i < n) {
        float d = deg[i];
        deg[i] = (d > 0.0f) ? rsqrtf(d) : 0.0f;
    }
}

// ---------------------------------------------------------------------------
// WMMA GEMM: C[M x NCOL] = A[M x K] (row major, lda = LDA) * B[K x NCOL]
// One wave computes MT vertically-stacked 16x16 tiles of C (same 16 columns)
// via V_WMMA_F32_16X16X4_F32, so the B operand registers are reused MT times
// per k-step. ntile = blockIdx.y * wavesPerBlock + waveId.
// RELU fuses max(0, a) on the A operand (activation of the previous layer).
// GUARD=false requires M % (16*MT) == 0 (checked by the host launcher).
// ---------------------------------------------------------------------------
template <int K, int LDA, int NCOL, bool RELU, int MT, bool GUARD>
__global__ void gemm_wmma_f32_k(const float* __restrict__ A,
                                const float* __restrict__ B,
                                float* __restrict__ C, int M) {
    const int lane = threadIdx.x & 31;
    const int wave = threadIdx.x >> 5;
    const int wavesPerBlock = blockDim.x >> 5;
    const int n0 = (blockIdx.y * wavesPerBlock + wave) * 16;
    if (n0 >= NCOL) return;                 // wave-uniform exit: EXEC stays full

    const int m0   = blockIdx.x * (16 * MT);
    const int row  = lane & 15;             // M within tile (both half-waves)
    const int col  = lane & 15;             // N within tile
    const int koff = (lane >> 4) << 1;      // 0 for lanes 0-15, 2 for 16-31

    // Per-tile A row base (clamped only in the guarded fallback)
    size_t abase[MT];
#pragma unroll
    for (int t = 0; t < MT; ++t) {
        int mrow = m0 + t * 16 + row;
        if (GUARD && mrow >= M) mrow = M - 1;
        abase[t] = (size_t)mrow * LDA + koff;
    }

    v8f acc[MT];
#pragma unroll
    for (int t = 0; t < MT; ++t)
        acc[t] = (v8f){0.f, 0.f, 0.f, 0.f, 0.f, 0.f, 0.f, 0.f};

    for (int k0 = 0; k0 < K; k0 += 4) {
        // B 4x16 f32 layout: rows striped across lanes within a VGPR
        v2f b;
        b[0] = B[(size_t)(k0 + koff)     * NCOL + n0 + col];
        b[1] = B[(size_t)(k0 + koff + 1) * NCOL + n0 + col];
#pragma unroll
        for (int t = 0; t < MT; ++t) {
            // A 16x4 f32: lanes 0-15 hold (K=0,K=1), lanes 16-31 (K=2,K=3)
            v2f a = *reinterpret_cast<const v2f*>(A + abase[t] + k0);
            if (RELU) {
                a[0] = fmaxf(a[0], 0.0f);
                a[1] = fmaxf(a[1], 0.0f);
            }
            acc[t] = __builtin_amdgcn_wmma_f32_16x16x4_f32(
                /*neg_a=*/false, a, /*neg_b=*/false, b,
                /*c_mod=*/(short)0, acc[t], /*reuse_a=*/false, /*reuse_b=*/false);
        }
    }

    // C/D 16x16 f32 layout: VGPR r holds M=r (lanes 0-15) and M=r+8 (16-31)
    const int rbase = (lane >> 4) << 3;
#pragma unroll
    for (int t = 0; t < MT; ++t) {
#pragma unroll
        for (int r = 0; r < 8; ++r) {
            int m = m0 + t * 16 + r + rbase;
            if (!GUARD || m < M)
                C[(size_t)m * NCOL + n0 + col] = acc[t][r];
        }
    }
}

// ---------------------------------------------------------------------------
// GCN aggregation: h[i] = dinv[i]^2 * xw[i] + bias  (self loop + bias init),
// then wave-per-edge atomic scatter of dinv[d]*dinv[s]*xw[s].
// ---------------------------------------------------------------------------
__global__ void agg_init_k(const float* __restrict__ xw,
                           const float* __restrict__ dinv,
                           const float* __restrict__ bias,
                           float* __restrict__ h, int n, int F) {
    int idx = blockIdx.x * blockDim.x + threadIdx.x;
    int total = n * F;
    if (idx < total) {
        int i = idx / F;
        int f = idx - i * F;
        float di = dinv[i];
        h[idx] = xw[idx] * di * di + bias[f];
    }
}

template <int F>
__global__ void agg_edge_k(const int* __restrict__ src,
                           const int* __restrict__ dst, int e,
                           const float* __restrict__ xw,
                           const float* __restrict__ dinv,
                           float* __restrict__ h) {
    int wid  = (blockIdx.x * blockDim.x + threadIdx.x) >> 5;
    int lane = threadIdx.x & 31;
    if (wid >= e) return;
    int s = src[wid];
    int d = dst[wid];
    float w = dinv[s] * dinv[d];
    const float* xs = xw + (size_t)s * F;
    float*       hd = h  + (size_t)d * F;
    if constexpr (F == 64) {
        // 2 floats per lane: one b64 load, two float atomics
        float2 v = *reinterpret_cast<const float2*>(xs + lane * 2);
        atomicAdd(&hd[lane * 2 + 0], v.x * w);
        atomicAdd(&hd[lane * 2 + 1], v.y * w);
    } else {
        // 8 floats per lane: two b128 loads, eight float atomics
        float4 v0 = *reinterpret_cast<const float4*>(xs + lane * 8);
        float4 v1 = *reinterpret_cast<const float4*>(xs + lane * 8 + 4);
        atomicAdd(&hd[lane * 8 + 0], v0.x * w);
        atomicAdd(&hd[lane * 8 + 1], v0.y * w);
        atomicAdd(&hd[lane * 8 + 2], v0.z * w);
        atomicAdd(&hd[lane * 8 + 3], v0.w * w);
        atomicAdd(&hd[lane * 8 + 4], v1.x * w);
        atomicAdd(&hd[lane * 8 + 5], v1.y * w);
        atomicAdd(&hd[lane * 8 + 6], v1.z * w);
        atomicAdd(&hd[lane * 8 + 7], v1.w * w);
    }
}

// ---------------------------------------------------------------------------
// Layer 3: zpre[i] = sum_k relu(h2[i,k]) * W3[k]   (N-dim is 1 -> wave dot)
// ---------------------------------------------------------------------------
__global__ void dot_w3_k(const float* __restrict__ h2,
                         const float* __restrict__ w3,
                         float* __restrict__ zpre, int n) {
    int wid  = (blockIdx.x * blockDim.x + threadIdx.x) >> 5;
    int lane = threadIdx.x & 31;
    if (wid >= n) return;
    const float* r = h2 + (size_t)wid * H2 + lane * 8;
    const float* wv = w3 + lane * 8;
    float4 a0 = *reinterpret_cast<const float4*>(r);
    float4 a1 = *reinterpret_cast<const float4*>(r + 4);
    float4 w0 = *reinterpret_cast<const float4*>(wv);
    float4 w1 = *reinterpret_cast<const float4*>(wv + 4);
    float s = fmaxf(a0.x, 0.f) * w0.x + fmaxf(a0.y, 0.f) * w0.y +
              fmaxf(a0.z, 0.f) * w0.z + fmaxf(a0.w, 0.f) * w0.w +
              fmaxf(a1.x, 0.f) * w1.x + fmaxf(a1.y, 0.f) * w1.y +
              fmaxf(a1.z, 0.f) * w1.z + fmaxf(a1.w, 0.f) * w1.w;
    for (int off = 16; off; off >>= 1)
        s += __shfl_xor(s, off, 32);
    if (lane == 0) zpre[wid] = s;
}

__global__ void z_init_k(const float* __restrict__ zpre,
                         const float* __restrict__ dinv,
                         const float* __restrict__ b3,
                         float* __restrict__ z, int n) {
    int i = blockIdx.x * blockDim.x + threadIdx.x;
    if (i < n) {
        float di = dinv[i];
        z[i] = zpre[i] * di * di + b3[0];
    }
}

__global__ void z_edge_k(const int* __restrict__ src,
                         const int* __restrict__ dst, int e,
                         const float* __restrict__ zpre,
                         const float* __restrict__ dinv,
                         float* __restrict__ z) {
    int i = blockIdx.x * blockDim.x + threadIdx.x;
    if (i < e) {
        int s = src[i], d = dst[i];
        atomicAdd(&z[d], zpre[s] * dinv[s] * dinv[d]);
    }
}

// ---------------------------------------------------------------------------
// Loss: mean over nodes of  softplus(-z) + (1-y)*z   (stable BCE-with-logits)
// ---------------------------------------------------------------------------
__global__ void zero_out_k(float* out) { out[0] = 0.0f; }

__global__ void bce_loss_k(const float* __restrict__ z,
                           const float* __restrict__ y,
                           float* __restrict__ out, int n) {
    int i = blockIdx.x * blockDim.x + threadIdx.x;
    float l = 0.0f;
    if (i < n) {
        float zv = z[i];
        float yv = y[i];
        float sp = fmaxf(-zv, 0.0f) + log1pf(expf(-fabsf(zv)));
        l = (sp + (1.0f - yv) * zv) * (1.0f / (float)n);
    }
    for (int off = 16; off; off >>= 1)
        l += __shfl_xor(l, off, 32);
    __shared__ float wsum[8];
    int lane = threadIdx.x & 31;
    int wave = threadIdx.x >> 5;
    if (lane == 0) wsum[wave] = l;
    __syncthreads();
    if (wave == 0) {
        float v = (lane < (int)(blockDim.x >> 5)) ? wsum[lane] : 0.0f;
        for (int off = 16; off; off >>= 1)
            v += __shfl_xor(v, off, 32);
        if (lane == 0) atomicAdd(out, v);
    }
}

// ---------------------------------------------------------------------------
// Launcher
// ---------------------------------------------------------------------------
extern "C" void kernel_launch(void* const* d_in, const int* in_sizes, int n_in,
                              void* d_out, int out_size, void* d_ws, size_t ws_size,
                              hipStream_t stream) {
    (void)n_in; (void)out_size; (void)ws_size;

    const float* x  = (const float*)d_in[0];
    const int*   ei = (const int*)  d_in[1];
    const float* y  = (const float*)d_in[2];
    const float* W1 = (const float*)d_in[3];
    const float* b1 = (const float*)d_in[4];
    const float* W2 = (const float*)d_in[5];
    const float* b2 = (const float*)d_in[6];
    const float* W3 = (const float*)d_in[7];
    const float* b3 = (const float*)d_in[8];

    const int n = in_sizes[0] / F_IN;   // 100000
    const int e = in_sizes[1] / 2;      // 1600000
    const int* src = ei;
    const int* dst = ei + e;

    // Workspace layout (floats). All offsets are multiples of 16 bytes.
    float* ws   = (float*)d_ws;
    float* dinv = ws;                               // n
    float* xw1  = dinv + n;                         // n*H1
    float* h1   = xw1 + (size_t)n * H1;             // n*H1
    float* xw2  = h1  + (size_t)n * H1;             // n*H2
    float* h2   = xw2 + (size_t)n * H2;             // n*H2
    float* zpre = h2  + (size_t)n * H2;             // n
    float* zbuf = zpre + n;                         // n

    float* out = (float*)d_out;

    const int T = 256;
    const int nb  = (n + T - 1) / T;
    const int eb  = (e + T - 1) / T;
    const int ewb = (e + 7) / 8;        // wave-per-edge, 8 waves / block
    const int nwb = (n + 7) / 8;        // wave-per-node

    // ---- normalization ----
    deg_init_k<<<nb, T, 0, stream>>>(dinv, n);
    deg_count_k<<<eb, T, 0, stream>>>(dst, e, dinv);
    deg_rsqrt_k<<<nb, T, 0, stream>>>(dinv, n);

    const bool mt5 = (n % 80) == 0;     // true for N=100000

    // ---- layer 1: xw1 = x @ W1 ; h1 = aggregate(xw1) + b1 ----
    if (mt5) {
        dim3 grid(n / 80, 1);           // 4 waves cover NCOL=64
        gemm_wmma_f32_k<F_IN, F_IN, H1, false, 5, false>
            <<<grid, 128, 0, stream>>>(x, W1, xw1, n);
    } else {
        dim3 grid((n + 15) / 16, 1);
        gemm_wmma_f32_k<F_IN, F_IN, H1, false, 1, true>
            <<<grid, 128, 0, stream>>>(x, W1, xw1, n);
    }
    agg_init_k<<<((n * H1) + T - 1) / T, T, 0, stream>>>(xw1, dinv, b1, h1, n, H1);
    agg_edge_k<H1><<<ewb, T, 0, stream>>>(src, dst, e, xw1, dinv, h1);

    // ---- layer 2: xw2 = relu(h1) @ W2 ; h2 = aggregate(xw2) + b2 ----
    if (mt5) {
        dim3 grid(n / 80, 2);           // 2 * 8 waves = 16 ntiles = 256 cols
        gemm_wmma_f32_k<H1, H1, H2, true, 5, false>
            <<<grid, 256, 0, stream>>>(h1, W2, xw2, n);
    } else {
        dim3 grid((n + 15) / 16, 2);
        gemm_wmma_f32_k<H1, H1, H2, true, 1, true>
            <<<grid, 256, 0, stream>>>(h1, W2, xw2, n);
    }
    agg_init_k<<<((n * H2) + T - 1) / T, T, 0, stream>>>(xw2, dinv, b2, h2, n, H2);
    agg_edge_k<H2><<<ewb, T, 0, stream>>>(src, dst, e, xw2, dinv, h2);

    // ---- layer 3: zpre = relu(h2) @ W3 ; z = aggregate(zpre) + b3 ----
    dot_w3_k<<<nwb, T, 0, stream>>>(h2, W3, zpre, n);
    z_init_k<<<nb, T, 0, stream>>>(zpre, dinv, b3, zbuf, n);
    z_edge_k<<<eb, T, 0, stream>>>(src, dst, e, zpre, dinv, zbuf);

    // ---- loss ----
    zero_out_k<<<1, 1, 0, stream>>>(out);
    bce_loss_k<<<nb, T, 0, stream>>>(zbuf, y, out, n);
}